// PerceptualMetric_43568148250949
// MI455X (gfx1250) — compile-verified
//
#include <hip/hip_runtime.h>

typedef __attribute__((ext_vector_type(2))) float v2f;
typedef __attribute__((ext_vector_type(8))) float v8f;

#define N_X   1024   // rows of x  (columns of out)
#define M_Y   1024   // rows of y  (rows of out)
#define D_DIM 512    // D_IN == D_FEAT

// ---------------------------------------------------------------------------
// f32 WMMA fragment layouts (per CDNA5 ISA 7.12.2):
//  A (16x4):  lane l<16  -> {A[M=l][k+0],  A[M=l][k+1]}
//             lane l>=16 -> {A[M=l-16][k+2], A[M=l-16][k+3]}
//  B (4x16):  lane l<16  -> {B[k+0][N=l],  B[k+1][N=l]}
//             lane l>=16 -> {B[k+2][N=l-16], B[k+3][N=l-16]}
//  C/D (16x16, 8 VGPRs): lane l<16: vgpr r = D[M=r][N=l]
//                        lane l>=16: vgpr r = D[M=8+r][N=l-16]
//
// Each wave computes a 32x16 output tile: two M-subtiles share every B
// fragment (halves B traffic) and give 4 independent WMMA accumulator
// chains (hides WMMA->WMMA RAW latency on the f32 matrix pipe).
// ---------------------------------------------------------------------------

// Kernel 1: F = [x;y] @ W   (2048x512 = 2048x512 * 512x512)
__global__ __launch_bounds__(256) void pm_project(const float* __restrict__ x,
                                                  const float* __restrict__ y,
                                                  const float* __restrict__ W,
                                                  float* __restrict__ F) {
  const int wave = blockIdx.x * (blockDim.x >> 5) + (threadIdx.x >> 5);
  const int lane = threadIdx.x & 31;
  const int half = lane >> 4;       // 0: K pair {0,1}, 1: K pair {2,3}
  const int l    = lane & 15;

  const int tilesN = D_DIM / 16;    // 32 column tiles
  const int tm = wave / tilesN;     // 0..63  (2048/32 row tiles over [x;y])
  const int tn = wave % tilesN;

  const int arow0 = tm * 32 + l;
  const int arow1 = arow0 + 16;
  const float* __restrict__ srow0 =
      (arow0 < N_X) ? (x + (size_t)arow0 * D_DIM)
                    : (y + (size_t)(arow0 - N_X) * D_DIM);
  const float* __restrict__ srow1 =
      (arow1 < N_X) ? (x + (size_t)arow1 * D_DIM)
                    : (y + (size_t)(arow1 - N_X) * D_DIM);
  const int bcol = tn * 16 + l;

  v8f c00 = {}, c01 = {};           // M-subtile 0, K-phase 0/1
  v8f c10 = {}, c11 = {};           // M-subtile 1, K-phase 0/1
  for (int k = 0; k < D_DIM; k += 8) {
    const int ka = k + 2 * half;
    v2f a00 = *(const v2f*)(srow0 + ka);
    v2f a01 = *(const v2f*)(srow0 + ka + 4);
    v2f a10 = *(const v2f*)(srow1 + ka);
    v2f a11 = *(const v2f*)(srow1 + ka + 4);
    v2f b0, b1;                     // W columns: stride-D rows, shared by both M-subtiles
    b0.x = W[(size_t)(ka + 0) * D_DIM + bcol];
    b0.y = W[(size_t)(ka + 1) * D_DIM + bcol];
    b1.x = W[(size_t)(ka + 4) * D_DIM + bcol];
    b1.y = W[(size_t)(ka + 5) * D_DIM + bcol];
    c00 = __builtin_amdgcn_wmma_f32_16x16x4_f32(false, a00, false, b0,
                                                (short)0, c00, false, false);
    c10 = __builtin_amdgcn_wmma_f32_16x16x4_f32(false, a10, false, b0,
                                                (short)0, c10, false, false);
    c01 = __builtin_amdgcn_wmma_f32_16x16x4_f32(false, a01, false, b1,
                                                (short)0, c01, false, false);
    c11 = __builtin_amdgcn_wmma_f32_16x16x4_f32(false, a11, false, b1,
                                                (short)0, c11, false, false);
  }

  const int col = tn * 16 + l;
#pragma unroll
  for (int r = 0; r < 8; ++r) {
    const int rr   = half ? (8 + r) : r;
    const int row0 = tm * 32 + rr;
    const int row1 = row0 + 16;
    F[(size_t)row0 * D_DIM + col] = c00[r] + c01[r];
    F[(size_t)row1 * D_DIM + col] = c10[r] + c11[r];
  }
}

// ---------------------------------------------------------------------------
// Kernel 2: sq[i] = sum_f F[i][f]^2, i in [0, 2048). One wave32 per row,
// coalesced strided reads + xor-shuffle tree reduction.
// ---------------------------------------------------------------------------
__global__ __launch_bounds__(128) void pm_sqnorm(const float* __restrict__ F,
                                                 float* __restrict__ sq) {
  const int row  = blockIdx.x * (blockDim.x >> 5) + (threadIdx.x >> 5);
  const int lane = threadIdx.x & 31;
  const float* __restrict__ p = F + (size_t)row * D_DIM;
  float s = 0.f;
#pragma unroll 4
  for (int j = lane; j < D_DIM; j += 32) {
    const float v = p[j];
    s += v * v;
  }
#pragma unroll
  for (int off = 16; off; off >>= 1) s += __shfl_xor(s, off, 32);
  if (lane == 0) sq[row] = s;
}

// ---------------------------------------------------------------------------
// Kernel 3: out[m,n] = sqrt(max(0, ny[m] + nx[n] - 2 * (fy @ fxT)[m,n])).
// A = fy rows; B[k][n] = fx[n][k] -> both fragments are contiguous float2
// loads from row-major data. 32x16 tile per wave; B fragments shared by the
// two M-subtiles. Norm-combine + sqrt fused into the epilogue: the [M,N,D]
// broadcast intermediate never exists.
// ---------------------------------------------------------------------------
__global__ __launch_bounds__(256) void pm_pairdist(const float* __restrict__ F,
                                                   const float* __restrict__ sq,
                                                   float* __restrict__ out) {
  const float* __restrict__ fx = F;
  const float* __restrict__ fy = F + (size_t)N_X * D_DIM;
  const float* __restrict__ nx = sq;
  const float* __restrict__ ny = sq + N_X;

  const int wave = blockIdx.x * (blockDim.x >> 5) + (threadIdx.x >> 5);
  const int lane = threadIdx.x & 31;
  const int half = lane >> 4;
  const int l    = lane & 15;

  const int tilesN = N_X / 16;      // 64
  const int tm = wave / tilesN;     // 0..31 (1024/32 row tiles over M)
  const int tn = wave % tilesN;     // column tile over N

  const float* __restrict__ arow0 = fy + (size_t)(tm * 32 + l) * D_DIM;
  const float* __restrict__ arow1 = arow0 + (size_t)16 * D_DIM;
  const float* __restrict__ brow  = fx + (size_t)(tn * 16 + l) * D_DIM;

  v8f c00 = {}, c01 = {};
  v8f c10 = {}, c11 = {};
  for (int k = 0; k < D_DIM; k += 8) {
    const int ka = k + 2 * half;
    v2f a00 = *(const v2f*)(arow0 + ka);
    v2f a01 = *(const v2f*)(arow0 + ka + 4);
    v2f a10 = *(const v2f*)(arow1 + ka);
    v2f a11 = *(const v2f*)(arow1 + ka + 4);
    v2f b0  = *(const v2f*)(brow + ka);
    v2f b1  = *(const v2f*)(brow + ka + 4);
    c00 = __builtin_amdgcn_wmma_f32_16x16x4_f32(false, a00, false, b0,
                                                (short)0, c00, false, false);
    c10 = __builtin_amdgcn_wmma_f32_16x16x4_f32(false, a10, false, b0,
                                                (short)0, c10, false, false);
    c01 = __builtin_amdgcn_wmma_f32_16x16x4_f32(false, a01, false, b1,
                                                (short)0, c01, false, false);
    c11 = __builtin_amdgcn_wmma_f32_16x16x4_f32(false, a11, false, b1,
                                                (short)0, c11, false, false);
  }

  const int n   = tn * 16 + l;
  const float xn = nx[n];
#pragma unroll
  for (int r = 0; r < 8; ++r) {
    const int rr = half ? (8 + r) : r;
    const int m0 = tm * 32 + rr;
    const int m1 = m0 + 16;
    const float g0  = c00[r] + c01[r];
    const float g1  = c10[r] + c11[r];
    const float d20 = ny[m0] + xn - 2.0f * g0;
    const float d21 = ny[m1] + xn - 2.0f * g1;
    out[(size_t)m0 * N_X + n] = sqrtf(fmaxf(d20, 0.0f));
    out[(size_t)m1 * N_X + n] = sqrtf(fmaxf(d21, 0.0f));
  }
}

// ---------------------------------------------------------------------------
extern "C" void kernel_launch(void* const* d_in, const int* in_sizes, int n_in,
                              void* d_out, int out_size, void* d_ws, size_t ws_size,
                              hipStream_t stream) {
  (void)in_sizes; (void)n_in; (void)out_size; (void)ws_size;
  const float* x = (const float*)d_in[0];   // [1024, 512]
  const float* y = (const float*)d_in[1];   // [1024, 512]
  const float* W = (const float*)d_in[2];   // [512, 512]
  // d_in[3] = dim (always the feature axis) — ignored.
  float* out = (float*)d_out;               // [1024, 1024]

  // Workspace: F = [fx; fy] (2048*512 f32 = 4 MB), then sq (2048 f32).
  float* F  = (float*)d_ws;
  float* sq = F + (size_t)(N_X + M_Y) * D_DIM;

  // F = [x;y] @ W : (2048/32)*(512/16) = 2048 waves, 8 waves/block
  pm_project<<<256, 256, 0, stream>>>(x, y, W, F);

  // Row squared norms: 2048 rows, 4 waves/block
  pm_sqnorm<<<512, 128, 0, stream>>>(F, sq);

  // Pairwise distances: (1024/32)*(1024/16) = 2048 waves, 8 waves/block
  pm_pairdist<<<256, 256, 0, stream>>>(F, sq, out);
}